// EnvelopeFollower_30245159698452
// MI455X (gfx1250) — compile-verified
//
#include <hip/hip_runtime.h>

// Envelope follower: s' = max(ga*s + (1-ga)*|x|, gr*s + (1-gr)*|x|)
// (max-form is exactly equivalent to the ga/gr branch select since ga < gr)

typedef float v4f __attribute__((ext_vector_type(4)));

#define B_SZ   32
#define T_SZ   262144
#define L_CHUNK 1024                 // output chunk per thread (timesteps)
#define N_CHUNK (T_SZ / L_CHUNK)     // 256 chunks per (b) chain
#define WARM   24576                 // warmup steps: state error <= gr^WARM ~ exp(-5.6)
#define TILE_TS 16                   // timesteps per LDS tile per thread
#define TILE_F4 (TILE_TS / 2)        // 8 v4f per tile per thread
#define NBUF   3                     // triple buffer -> 2 tiles in flight
#define BLOCK_THREADS 256

#define GA_C 0.99773500f             // exp(-1/(44100*0.010))
#define GR_C 0.99977327f             // exp(-1/(44100*0.100))
#define CA_C 0.0022650047f           // 1 - GA
#define CR_C 0.00022673167f          // 1 - GR

__device__ __forceinline__ float env_step(float s, float ax) {
  // two independent FMAs (VOPD dual-issue friendly); chain: fma -> max
  return fmaxf(fmaf(GA_C, s, CA_C * ax), fmaf(GR_C, s, CR_C * ax));
}

__device__ __forceinline__ void async_load_f4(const v4f* g, const v4f* l) {
  // LDS generic pointers: low 32 bits are the LDS byte offset (aperture layout)
  unsigned lds = (unsigned)(unsigned long long)l;
  asm volatile("global_load_async_to_lds_b128 %0, %1, off"
               :: "v"(lds), "v"(g) : "memory");
}

__device__ __forceinline__ void wait_async_le16() {
  asm volatile("s_wait_asynccnt 0x10" ::: "memory");
}
__device__ __forceinline__ void wait_async_le8() {
  asm volatile("s_wait_asynccnt 0x8" ::: "memory");
}
__device__ __forceinline__ void wait_async_0() {
  asm volatile("s_wait_asynccnt 0x0" ::: "memory");
}

__device__ __forceinline__ void stage_tile(const v4f* __restrict__ gsrc, int tile,
                                           v4f (*buf)[BLOCK_THREADS], int tIdx) {
  const v4f* g = gsrc + (size_t)tile * TILE_F4;
#pragma unroll
  for (int j = 0; j < TILE_F4; ++j)
    async_load_f4(g + j, &buf[j][tIdx]);
}

__global__ __launch_bounds__(BLOCK_THREADS)
void EnvelopeFollower_kernel(const float* __restrict__ in, float* __restrict__ out) {
  // LDS tile layout [buf][j][tid]: lane-major => ds_load_b128 bank-conflict free
  __shared__ v4f smem[NBUF][TILE_F4][BLOCK_THREADS];

  const int tIdx = threadIdx.x;
  const int gtid = blockIdx.x * BLOCK_THREADS + tIdx;
  const int b    = gtid & 31;     // lane id == batch index -> uniform k per wave
  const int k    = gtid >> 5;     // chunk id (uniform across the wave32)

  const int t0        = k * L_CHUNK;
  const int start     = (t0 >= WARM) ? (t0 - WARM) : 0;  // exact from t=0 for early chunks
  const int nTiles    = (t0 + L_CHUNK - start) / TILE_TS; // uniform per wave (>= 64)
  const int warmTiles = (t0 - start) / TILE_TS;

  // v4f = {x(t,c0), x(t,c1), x(t+1,c0), x(t+1,c1)}; start is even -> 16B aligned
  const v4f* gsrc = reinterpret_cast<const v4f*>(in)
                  + (((size_t)b * T_SZ + (size_t)start) >> 1);
  v4f* gdst = reinterpret_cast<v4f*>(out)
            + (((size_t)b * T_SZ + (size_t)t0) >> 1);

  // Prologue: two tiles in flight
  stage_tile(gsrc, 0, smem[0], tIdx);
  stage_tile(gsrc, 1, smem[1], tIdx);

  float s0 = 0.0f, s1 = 0.0f;
  int outIdx = 0;
  int bufc = 0;

  for (int tile = 0; tile < nTiles; ++tile) {
    // issue tile+2 into the buffer consumed one iteration ago, then wait for
    // the current tile (async loads complete in order):
    //   2 newer tiles outstanding -> cnt<=16; 1 newer -> cnt<=8; none -> 0
    if (tile + 2 < nTiles) {
      int bnext = bufc + 2; if (bnext >= NBUF) bnext -= NBUF;
      stage_tile(gsrc, tile + 2, smem[bnext], tIdx);
      wait_async_le16();
    } else if (tile + 1 < nTiles) {
      wait_async_le8();
    } else {
      wait_async_0();
    }

    const bool emit = (tile >= warmTiles);  // uniform per wave
#pragma unroll
    for (int j = 0; j < TILE_F4; ++j) {
      v4f v = smem[bufc][j][tIdx];
      float e0 = env_step(s0, fabsf(v.x));
      float e1 = env_step(s1, fabsf(v.y));
      float e2 = env_step(e0, fabsf(v.z));
      float e3 = env_step(e1, fabsf(v.w));
      s0 = e2; s1 = e3;
      if (emit) {
        v4f r = {e0, e1, e2, e3};
        // output is write-once: NT store keeps the 25x re-read input L2-resident
        __builtin_nontemporal_store(r, &gdst[outIdx + j]);
      }
    }
    if (emit) outIdx += TILE_F4;
    ++bufc; if (bufc == NBUF) bufc = 0;
  }
}

extern "C" void kernel_launch(void* const* d_in, const int* in_sizes, int n_in,
                              void* d_out, int out_size, void* d_ws, size_t ws_size,
                              hipStream_t stream) {
  (void)in_sizes; (void)n_in; (void)out_size; (void)d_ws; (void)ws_size;
  const float* in = (const float*)d_in[0];
  float* out = (float*)d_out;
  const int totalThreads = B_SZ * N_CHUNK;           // 8192 threads = 256 waves
  dim3 grid(totalThreads / BLOCK_THREADS);           // 32 blocks
  dim3 block(BLOCK_THREADS);
  EnvelopeFollower_kernel<<<grid, block, 0, stream>>>(in, out);
}